// SNNModel0HLayers_38826504355948
// MI455X (gfx1250) — compile-verified
//
#include <hip/hip_runtime.h>

typedef __attribute__((ext_vector_type(2))) float v2f;
typedef __attribute__((ext_vector_type(8))) float v8f;

#define T_DIM 200
#define F_DIM 128
#define B_DIM 4096
#define O_DIM 2
#define WAVES_PER_BLOCK 4

// One wave32 per batch element b.
//   cur^T[o, t] = sum_f W[o,f] * x[b,f,t] + bias[o]
// WMMA f32 16x16x4:  A = W tile (MxK = 16x4, rows o=0,1 valid, rest zero)
//                    B = x tile (KxN = 4x16, N = 16 consecutive timesteps)
// K=128 -> 32 WMMAs per 16-timestep tile (two independent accumulators).
// Then a sequential LIF scan over the 16 timesteps (all lanes redundantly,
// via shfl broadcast), lane j records spike for t0+j, coalesced store.
__global__ __launch_bounds__(128) void snn_lif_wmma_kernel(
    const float* __restrict__ x,      // [B, F, T]
    const float* __restrict__ W,      // [O, F]
    const float* __restrict__ bias,   // [O]
    const float* __restrict__ beta_p, // [1]
    float* __restrict__ out)          // [T, B, O]
{
  const int lane = threadIdx.x & 31;
  const int wave = threadIdx.x >> 5;
  const int b    = blockIdx.x * WAVES_PER_BLOCK + wave;

  const float beta  = beta_p[0];
  const float bias0 = bias[0];
  const float bias1 = bias[1];

  const int m     = lane & 15;        // M index (A) / N index (B) within half
  const int khalf = (lane >> 4) * 2;  // lanes 0-15 -> K pair {0,1}; 16-31 -> {2,3}

  // Preload all 32 A fragments of W (16x4 f32 layout, ISA 7.12.2):
  //   lanes 0-15:  V0 = A[m, f0+0], V1 = A[m, f0+1]
  //   lanes 16-31: V0 = A[m, f0+2], V1 = A[m, f0+3]
  v2f afrag[32];
#pragma unroll
  for (int kk = 0; kk < 32; ++kk) {
    const int f0 = kk * 4 + khalf;
    float ax = 0.0f, ay = 0.0f;
    if (m < O_DIM) {
      ax = W[m * F_DIM + f0 + 0];
      ay = W[m * F_DIM + f0 + 1];
    }
    afrag[kk].x = ax;
    afrag[kk].y = ay;
  }

  const float* xb = x + (size_t)b * (F_DIM * T_DIM);

  // LIF state (replicated identically across all lanes of the wave)
  float mem0 = 0.0f, mem1 = 0.0f, spk0 = 0.0f, spk1 = 0.0f;

  for (int t0 = 0; t0 < T_DIM; t0 += 16) {
    // Two accumulators to break the WMMA RAW chain; bias seeded into rows
    // M=0 / M=1 of C (VGPR v: lanes 0-15 hold M=v, N=lane).
    v8f ca = {};
    v8f cb = {};
    if (lane < 16) { ca[0] = bias0; ca[1] = bias1; }

    const int tl = min(t0 + m, T_DIM - 1); // clamp tail tile (results unused)

#pragma unroll
    for (int kk = 0; kk < 32; kk += 2) {
      const int f0 = kk * 4 + khalf;
      const int f1 = (kk + 1) * 4 + khalf;
      v2f bfa, bfb;
      bfa.x = xb[(f0 + 0) * T_DIM + tl];   // 16 consecutive t's across lanes
      bfa.y = xb[(f0 + 1) * T_DIM + tl];
      bfb.x = xb[(f1 + 0) * T_DIM + tl];
      bfb.y = xb[(f1 + 1) * T_DIM + tl];
      ca = __builtin_amdgcn_wmma_f32_16x16x4_f32(false, afrag[kk],     false, bfa,
                                                 (short)0, ca, false, false);
      cb = __builtin_amdgcn_wmma_f32_16x16x4_f32(false, afrag[kk + 1], false, bfb,
                                                 (short)0, cb, false, false);
    }

    // lanes 0-15: cur(o, t0+lane)
    const float cur0 = ca[0] + cb[0];
    const float cur1 = ca[1] + cb[1];

    float rec0 = 0.0f, rec1 = 0.0f;
    const int jmax = min(16, T_DIM - t0);
    for (int j = 0; j < jmax; ++j) {
      const float c0 = __shfl(cur0, j, 32);
      const float c1 = __shfl(cur1, j, 32);
      // snntorch Leaky, reset-by-subtract with previous-step spike
      mem0 = beta * mem0 + c0 - spk0;   // THRESHOLD = 1.0
      mem1 = beta * mem1 + c1 - spk1;
      spk0 = (mem0 > 1.0f) ? 1.0f : 0.0f;
      spk1 = (mem1 > 1.0f) ? 1.0f : 0.0f;
      rec0 = (lane == j) ? spk0 : rec0;
      rec1 = (lane == j) ? spk1 : rec1;
    }

    // lane j stores spikes for t = t0 + j (8B per lane; merged in L2)
    if (lane < jmax) {
      float* o = out + (size_t)(t0 + lane) * (B_DIM * O_DIM) + b * O_DIM;
      o[0] = rec0;
      o[1] = rec1;
    }
  }
}

extern "C" void kernel_launch(void* const* d_in, const int* in_sizes, int n_in,
                              void* d_out, int out_size, void* d_ws, size_t ws_size,
                              hipStream_t stream) {
  const float* x    = (const float*)d_in[0];   // [4096, 128, 200]
  const float* W    = (const float*)d_in[1];   // [2, 128]
  const float* bias = (const float*)d_in[2];   // [2]
  const float* beta = (const float*)d_in[3];   // [1]
  float* out = (float*)d_out;                  // [200, 4096, 2]

  dim3 grid(B_DIM / WAVES_PER_BLOCK);
  dim3 block(32 * WAVES_PER_BLOCK);
  hipLaunchKernelGGL(snn_lif_wmma_kernel, grid, block, 0, stream,
                     x, W, bias, beta, out);
}